// RNN_3PG_50379966382619
// MI455X (gfx1250) — compile-verified
//
#include <hip/hip_runtime.h>
#include <stdint.h>

// ---------------------------------------------------------------------------
// 3-PG forest growth RNN for MI455X (gfx1250).
// One thread = one site (batch element). State lives in VGPRs for all 120
// steps. Inputs are staged global->LDS with CDNA5 async copies
// (global_load_async_to_lds_b32, ASYNCcnt) in a double buffer so the 9120B
// lane-strided HBM reads are always one step ahead of the VALU work.
// ---------------------------------------------------------------------------

#define TSTEPS   120
#define XCOLS    19
#define NLOAD    16   // columns 2..17 are the only ones consumed
#define LSTRIDE  17   // padded LDS slot stride in floats (gcd(17,64)=1 -> no bank conflicts)
#define BLK      256
#define NPARAM   42

// ---- CDNA5 async global->LDS copy (per-lane, tracked by ASYNCcnt) ----------
// GVS addressing: mem = SADDR(64b sgpr) + VADDR(32b vgpr) + offset
// The immediate offset is added to BOTH the LDS and global address
// (CDNA5 ISA), so consecutive floats of a row land in
// consecutive floats of the lane's LDS slot.
#define ASYNC_B32(lds, sbase, voff, IMM)                                     \
  asm volatile("global_load_async_to_lds_b32 %0, %1, %2 offset:" #IMM        \
               :: "v"(lds), "v"(voff), "s"(sbase) : "memory")

__device__ __forceinline__ void issue_row16(uint32_t lds, uint64_t sbase, uint32_t voff) {
  ASYNC_B32(lds, sbase, voff, 0);
  ASYNC_B32(lds, sbase, voff, 4);
  ASYNC_B32(lds, sbase, voff, 8);
  ASYNC_B32(lds, sbase, voff, 12);
  ASYNC_B32(lds, sbase, voff, 16);
  ASYNC_B32(lds, sbase, voff, 20);
  ASYNC_B32(lds, sbase, voff, 24);
  ASYNC_B32(lds, sbase, voff, 28);
  ASYNC_B32(lds, sbase, voff, 32);
  ASYNC_B32(lds, sbase, voff, 36);
  ASYNC_B32(lds, sbase, voff, 40);
  ASYNC_B32(lds, sbase, voff, 44);
  ASYNC_B32(lds, sbase, voff, 48);
  ASYNC_B32(lds, sbase, voff, 52);
  ASYNC_B32(lds, sbase, voff, 56);
  ASYNC_B32(lds, sbase, voff, 60);
}

__device__ __forceinline__ void wait_async_le16() {
  asm volatile("s_wait_asynccnt 0x10" ::: "memory");
}
__device__ __forceinline__ void wait_async_0() {
  asm volatile("s_wait_asynccnt 0x0" ::: "memory");
}

// ---- raw gfx1250 transcendental-pipe ops ----------------------------------
// v_log_f32 (log2) and v_exp_f32 (2^x); these co-execute with normal VALU.
__device__ __forceinline__ float hw_log2(float x) { return __builtin_amdgcn_logf(x); }
__device__ __forceinline__ float hw_exp2(float x) { return __builtin_amdgcn_exp2f(x); }

// pow: x==0 with y>0 -> exp2(-inf) -> 0, matching jnp semantics for our uses.
__device__ __forceinline__ float fpow(float x, float y) {
  return hw_exp2(y * hw_log2(x));
}
// e^x via the same pipe
#define LOG2E 1.4426950408889634f
__device__ __forceinline__ float fexp(float x) { return hw_exp2(x * LOG2E); }

__device__ __forceinline__ float clamp01(float v) {
  return fminf(fmaxf(v, 0.0f), 1.0f);
}

// ---- parameter gathering ---------------------------------------------------
struct ParamPtrs { const float* p[NPARAM]; };

__global__ void pack_params_kernel(ParamPtrs pp, float* __restrict__ out) {
  int i = threadIdx.x;
  if (i < NPARAM) out[i] = pp.p[i][0];
}

// ---- main recurrence kernel ------------------------------------------------
__global__ __launch_bounds__(BLK) void rnn3pg_kernel(
    const float* __restrict__ x,    // [B,120,19]
    const float* __restrict__ h0,   // [B,13]
    const float* __restrict__ P,    // [42] packed params (dict order)
    float* __restrict__ out,        // [B,120,13]
    int B)
{
  __shared__ float xs[2][BLK * LSTRIDE];

  const int tid = threadIdx.x;
  const int b   = blockIdx.x * BLK + tid;
  if (b >= B) return;

  // ---- uniform params (dict insertion order) ----
  const float Tmin = P[0],  Topt = P[1],  Tmax = P[2],  kF = P[3];
  const float fN0  = P[4],  m0   = P[5],  nAge = P[6],  rAge = P[7];
  const float gammaFx = P[8], gammaF0 = P[9], tgammaF = P[10], Rttover = P[11];
  const float MaxCond = P[12], LAIgcx = P[13], CoeffCond = P[14], BLcond = P[15];
  const float wSx1000 = P[16], thinPower = P[17], mF = P[18], mR = P[19], mS = P[20];
  const float SLA0 = P[21], SLA1 = P[22], tSLA = P[23], kk = P[24];
  const float fullCanAge = P[25], alpha = P[26], yy = P[27];
  const float pRx = P[28], pRn = P[29], pFS2 = P[30], pFS20 = P[31];
  const float aWs = P[32], nWs = P[33], fracBB0 = P[34], fracBB1 = P[35];
  const float tBB = P[36], Density = P[37], MaxIntcptn = P[38];
  const float LAImaxIntcptn = P[39], Qa = P[40], Qb = P[41];

  // ---- loop-invariant derived constants ----
  const float LN2f        = 0.6931472f;
  const float SLAdiff     = SLA0 - SLA1;
  const float inv_tSLA    = 1.0f / tSLA;
  const float inv_TT      = 1.0f / (Topt - Tmin);
  const float inv_TmT     = 1.0f / (Tmax - Topt);
  const float fTexp       = (Tmax - Topt) * inv_TT;
  const float pfsPower    = hw_log2(pFS20 / pFS2) * 0.30102999566398f; // log10(x)=log2(x)*log10(2)
  const float pfsConst    = pFS2 / fpow(2.0f, pfsPower);
  const float gFk         = 12.0f * (hw_log2(1.0f + gammaFx / gammaF0) * LN2f) / tgammaF;
  const float gFxg0       = gammaFx * gammaF0;
  const float gFdiff      = gammaFx - gammaF0;
  const float inv_nWs     = 1.0f / nWs;
  const float inv_aWs     = 1.0f / aWs;
  const float inv_Density = 1.0f / Density;
  const float inv_tBB     = 1.0f / tBB;
  const float inv_fullCan = 1.0f / fullCanAge;
  const float inv_LAIgcx  = 1.0f / LAIgcx;
  const float inv_LAImaxI = 1.0f / LAImaxIntcptn;
  const float defConst    = 1.2f * 2460000.0f * 0.000622f * BLcond; // * VPD

  // ---- initial state (only the carried columns) ----
  const size_t hbase = (size_t)b * 13;
  float ASW    = h0[hbase + 2];
  float StemNoS= h0[hbase + 3];
  float age    = h0[hbase + 5];
  float WF     = h0[hbase + 6];
  float WR     = h0[hbase + 7];
  float WS     = h0[hbase + 8];
  float TotLit = h0[hbase + 9];
  float avDBHs = h0[hbase + 10];

  // ---- async staging setup ----
  const uint64_t sbase = (uint64_t)(uintptr_t)x;
  // byte offset of x[b, 0, col=2]
  uint32_t voff = (uint32_t)(((size_t)b * TSTEPS * XCOLS + 2) * sizeof(float));
  const uint32_t rowstride = XCOLS * sizeof(float);
  // low 32 bits of a generic LDS pointer == LDS byte offset (ISA aperture rule)
  const uint32_t lds0 = (uint32_t)(uintptr_t)(&xs[0][tid * LSTRIDE]);
  const uint32_t lds1 = (uint32_t)(uintptr_t)(&xs[1][tid * LSTRIDE]);

  issue_row16(lds0, sbase, voff);   // prefetch t = 0

  float* op = out + (size_t)b * TSTEPS * 13;

  for (int t = 0; t < TSTEPS; ++t) {
    // prefetch t+1 into the other buffer, then wait for the current buffer
    if (t + 1 < TSTEPS) {
      issue_row16((t & 1) ? lds0 : lds1, sbase, voff + rowstride);
      wait_async_le16();            // <=16 outstanding => current row landed
    } else {
      wait_async_0();
    }
    voff += rowstride;

    const float* xr = &xs[t & 1][tid * LSTRIDE];
    const float T_av   = xr[0];
    const float VPD    = xr[1];
    const float rain   = xr[2];
    const float solar  = xr[3];
    const float frost  = xr[5];
    const float Ca     = xr[6];
    const float D13Catm= xr[7];
    const float dayl   = xr[8];
    const float dim    = xr[9];
    const float MaxASW = xr[10];
    const float MinASW = xr[11];
    const float SWconst= xr[12];
    const float SWpower= xr[13];
    const float FR     = xr[14];
    const float MaxAge = xr[15];

    const float StemNo = fmaxf(StemNoS, 1.0f);
    const float avDBH  = fmaxf(avDBHs, 0.1f);

    // --- physiology ---
    const float ageR   = age * inv_tSLA;
    const float SLA    = SLA1 + SLAdiff * fexp(-LN2f * ageR * ageR);
    const float LAI    = WF * SLA * 0.1f;
    const float RADint = solar * dim;
    const float PAR    = RADint * 2.3f;

    const float bT1 = fmaxf((T_av - Tmin) * inv_TT, 0.0f);
    const float bT2 = fmaxf((Tmax - T_av) * inv_TmT, 0.0f);
    const float fT  = bT1 * fpow(bT2, fTexp);

    const float fFrost = 1.0f - kF * frost * (1.0f / 30.0f);
    const float fVPD   = fexp(-CoeffCond * VPD);
    const float moist  = clamp01(ASW / MaxASW);
    const float fSW    = 1.0f / (1.0f + fpow((1.0f - moist) / SWconst, SWpower));
    const float fNutr  = fN0 + (1.0f - fN0) * FR;
    const float fAge   = 1.0f / (1.0f + fpow(age / (rAge * MaxAge), nAge));
    const float PhysMod= fAge * fminf(fVPD, fSW);
    const float CanCover = clamp01((age + (1.0f / 12.0f)) * inv_fullCan);
    const float lightInt = 1.0f - fexp(-kk * fmaxf(LAI, 0.0f));
    const float alphaC = alpha * fNutr * fT * fFrost * PhysMod;
    const float GPPmolc= PAR * lightInt * CanCover * alphaC;
    const float NPP    = GPPmolc * 0.24f * yy;

    // --- allocation ---
    const float pFS = pfsConst * fpow(avDBH, pfsPower);
    const float m   = m0 + (1.0f - m0) * FR;
    const float pR  = pRx * pRn / (pRn + (pRx - pRn) * PhysMod * m);
    const float pS  = (1.0f - pR) / (1.0f + pFS);
    const float pF  = 1.0f - pR - pS;

    const float gammaF = gFxg0 / (gammaF0 + gFdiff * fexp(-gFk * age));
    const float litter = gammaF * WF;
    const float WF2 = WF + NPP * pF - litter;
    const float WR2 = WR + NPP * pR - Rttover * WR;
    const float WS2 = WS + NPP * pS;
    const float TotLit2 = TotLit + litter;

    // --- self-thinning ---
    const float wSmax  = wSx1000 * fpow(1000.0f / StemNo, thinPower);
    const float avStem = WS2 * 1000.0f / StemNo;
    const float delStems = (avStem > wSmax) ? StemNo * (avStem - wSmax) / avStem : 0.0f;
    const float dSoverN  = delStems / StemNo;
    const float WF3 = WF2 - mF * dSoverN * WF2;
    const float WR3 = WR2 - mR * dSoverN * WR2;
    const float WS3 = WS2 - mS * dSoverN * WS2;
    const float StemNo2 = StemNo - delStems;

    // --- water balance ---
    const float CanCond = fmaxf(MaxCond * PhysMod * clamp01(LAI * inv_LAIgcx), 1e-4f);
    const float netRad  = Qa + Qb * (solar * 1000000.0f / dayl);
    const float defTerm = defConst * VPD;
    const float divd    = 3.2f + BLcond / CanCond;
    const float transp  = dim * fmaxf((2.2f * netRad + defTerm) / divd, 0.0f)
                          * (1.0f / 2460000.0f) * dayl;
    const float interc  = MaxIntcptn * clamp01(LAI * inv_LAImaxI) * rain;
    const float ASW2    = fminf(fmaxf(ASW + rain - transp - interc, MinASW), MaxASW);

    // --- stand metrics ---
    const float avStem2 = WS3 * 1000.0f / fmaxf(StemNo2, 1.0f);
    const float avDBH2  = fpow(fmaxf(avStem2, 0.001f) * inv_aWs, inv_nWs);
    const float fracBB  = fracBB1 + (fracBB0 - fracBB1) * fexp(-LN2f * age * inv_tBB);
    const float StandVol2 = WS3 * (1.0f - fracBB) * inv_Density;
    const float LAI2    = WF3 * SLA * 0.1f;

    // --- isotope discrimination ---
    const float GcMol = CanCond * (1000.0f / 24.6f);
    const float assim = GPPmolc / (dim * dayl);
    const float CiCa  = clamp01(1.0f - assim / (0.625f * GcMol * Ca * 1e-6f));
    const float D13C2 = D13Catm - 4.4f - 22.6f * CiCa;
    const float age2  = age + dim * (1.0f / 365.0f);

    // --- emit state [13] ---
    float* o = op + (size_t)t * 13;
    o[0]  = StandVol2;
    o[1]  = LAI2;
    o[2]  = ASW2;
    o[3]  = StemNo2;
    o[4]  = PAR;
    o[5]  = age2;
    o[6]  = WF3;
    o[7]  = WR3;
    o[8]  = WS3;
    o[9]  = TotLit2;
    o[10] = avDBH2;
    o[11] = delStems;
    o[12] = D13C2;

    // --- carry ---
    ASW = ASW2; StemNoS = StemNo2; age = age2;
    WF = WF3; WR = WR3; WS = WS3; TotLit = TotLit2; avDBHs = avDBH2;
  }
}

// ---------------------------------------------------------------------------
extern "C" void kernel_launch(void* const* d_in, const int* in_sizes, int n_in,
                              void* d_out, int out_size, void* d_ws, size_t ws_size,
                              hipStream_t stream) {
  const float* x  = (const float*)d_in[0];   // [B,120,19]
  const float* h0 = (const float*)d_in[1];   // [B,13]
  float* P = (float*)d_ws;                   // [42] packed params

  if (n_in >= 2 + NPARAM) {
    // params arrived as 42 single-element arrays (dict order)
    ParamPtrs pp;
    for (int i = 0; i < NPARAM; ++i) pp.p[i] = (const float*)d_in[2 + i];
    pack_params_kernel<<<1, 64, 0, stream>>>(pp, P);
  } else {
    // params arrived as one contiguous float[42]
    (void)hipMemcpyAsync(P, d_in[2], NPARAM * sizeof(float),
                         hipMemcpyDeviceToDevice, stream);
  }

  const int B = in_sizes[1] / 13;
  const int grid = (B + BLK - 1) / BLK;
  rnn3pg_kernel<<<grid, BLK, 0, stream>>>(x, h0, P, (float*)d_out, B);
}